// ConcatCorrelationAdaptor_12850542150351
// MI455X (gfx1250) — compile-verified
//
#include <hip/hip_runtime.h>
#include <hip/hip_bf16.h>

// ---------------------------------------------------------------------------
// ConcatCorrelationAdaptor for MI455X (gfx1250, wave32, WMMA).
//   T=2, B=4, C=256, H=64, W=64, DG=4, Cpg=64, KS=3
//   level0: disp=4 stride=2 -> 81 corr chans ; level1: disp=2 stride=1 -> 25
// Heavy compute = deformable-conv implicit GEMM (Cout=256, K=2304, N=16384
// per level) in fp32 WMMA (V_WMMA_F32_16X16X4_F32) for reference precision.
//  * bilinear sampling: clamp indices + fold validity into weights -> no
//    EXEC-mask branches around the 4 corner gathers
//  * weights pre-transposed to [k9][klin][cout] so A-fragment loads are
//    lane-coalesced
//  * correlation LDS staging via gfx1250 async global->LDS (ASYNCcnt);
//    builtin params are address_space(1)/(3) int* per hipcc diagnostic
// ---------------------------------------------------------------------------

typedef __attribute__((ext_vector_type(2))) float v2f;
typedef __attribute__((ext_vector_type(8))) float v8f;

#define NB 4          // batch
#define NC 256        // channels
#define HH 64
#define WW 64
#define HWN 4096      // H*W
#define NCORR 106     // 81 + 25
#define OFFCH 72      // DG*KS*KS*2

#if __has_builtin(__builtin_amdgcn_global_load_async_to_lds_b32) && \
    __has_builtin(__builtin_amdgcn_s_wait_asynccnt)
#define CCA_ASYNC_LDS 1
#else
#define CCA_ASYNC_LDS 0
#endif

// global->LDS copy of one dword per lane; async (ASYNCcnt) when available.
__device__ __forceinline__ void cca_g2l_b32(const float* g, float* l) {
#if CCA_ASYNC_LDS
  typedef __attribute__((address_space(1))) int* as1i;   // prints as __device__ int*
  typedef __attribute__((address_space(3))) int* as3i;   // LDS
  __builtin_amdgcn_global_load_async_to_lds_b32(
      (as1i)(unsigned long long)(const void*)g,
      (as3i)(unsigned int)(unsigned long long)(const void*)l, 0, 0);
#else
  *l = *g;
#endif
}

__device__ __forceinline__ void cca_wait_async() {
#if CCA_ASYNC_LDS
  __builtin_amdgcn_s_wait_asynccnt(0);
#endif
}

// ---------------------------------------------------------------------------
// 1) Windowed correlation (cost volume).  8x8 pixel tile per 64-thread block,
//    LDS staging of f1 tile and haloed f2 tile (halo pre-zeroed once, only
//    in-range elements copied per chunk), WIN*WIN register accumulators.
//    corr[b, ch_base+k, h, w] = (1/256) * sum_c f1[c,h,w]*f2[c,h+dy*S,w+dx*S]
// ---------------------------------------------------------------------------
template <int WIN, int STRIDE, int CHB>
__global__ void cca_corr_kernel(const float* __restrict__ f1,
                                const float* __restrict__ f2,
                                float* __restrict__ corr, int ch_base) {
  constexpr int DISP = (WIN - 1) / 2;
  constexpr int PADR = DISP * STRIDE;
  constexpr int BT = 8 + 2 * PADR;

  __shared__ float a_s[CHB][8][8];
  __shared__ float b_s[CHB][BT][BT];

  const int tid = threadIdx.x;
  const int tile = blockIdx.x;              // 0..63
  const int b = blockIdx.y;
  const int th0 = (tile >> 3) << 3;
  const int tw0 = (tile & 7) << 3;
  const int ph = tid >> 3, pw = tid & 7;
  const int h = th0 + ph, w = tw0 + pw;

  const float* f1b = f1 + (size_t)b * NC * HWN;
  const float* f2b = f2 + (size_t)b * NC * HWN;

  // zero the halo region once: out-of-range slots are never overwritten
  float* bflat = &b_s[0][0][0];
  for (int i = tid; i < CHB * BT * BT; i += 64) bflat[i] = 0.0f;
  __syncthreads();

  float acc[WIN * WIN];
#pragma unroll
  for (int k = 0; k < WIN * WIN; ++k) acc[k] = 0.0f;

  for (int c0 = 0; c0 < NC; c0 += CHB) {
#pragma unroll
    for (int cc = 0; cc < CHB; ++cc)
      cca_g2l_b32(&f1b[(size_t)(c0 + cc) * HWN + h * WW + w],
                  &a_s[cc][ph][pw]);

    for (int i = tid; i < CHB * BT * BT; i += 64) {
      int cc = i / (BT * BT);
      int r = (i / BT) % BT;
      int cl = i % BT;
      int gh = th0 - PADR + r;
      int gw = tw0 - PADR + cl;
      if (gh >= 0 && gh < HH && gw >= 0 && gw < WW)
        cca_g2l_b32(&f2b[(size_t)(c0 + cc) * HWN + gh * WW + gw],
                    &b_s[cc][r][cl]);
    }
    cca_wait_async();
    __syncthreads();

#pragma unroll
    for (int cc = 0; cc < CHB; ++cc) {
      float av = a_s[cc][ph][pw];
#pragma unroll
      for (int dy = 0; dy < WIN; ++dy)
#pragma unroll
        for (int dx = 0; dx < WIN; ++dx)
          acc[dy * WIN + dx] =
              fmaf(av, b_s[cc][ph + dy * STRIDE][pw + dx * STRIDE],
                   acc[dy * WIN + dx]);
    }
    __syncthreads();
  }

  const float inv = 1.0f / 256.0f;
#pragma unroll
  for (int k = 0; k < WIN * WIN; ++k)
    corr[((size_t)(b * NCORR + ch_base + k)) * HWN + h * WW + w] = acc[k] * inv;
}

// ---------------------------------------------------------------------------
// 2) Offset projection: offs[lvl,b,o,hw] = sum_c corr[b,c,hw] * w_off_l[o,c]
// ---------------------------------------------------------------------------
__global__ void cca_offset_kernel(const float* __restrict__ corr,
                                  const float* __restrict__ w0,
                                  const float* __restrict__ w1,
                                  float* __restrict__ offs) {
  int t = blockIdx.x * 256 + threadIdx.x;
  int hw = t & (HWN - 1);
  int rest = t >> 12;
  int o = rest % OFFCH;
  rest /= OFFCH;
  int b = rest & 3;
  int lvl = rest >> 2;

  const float* wo = (lvl == 0 ? w0 : w1) + o * NCORR;
  const float* cp = corr + (size_t)b * NCORR * HWN + hw;
  float acc = 0.0f;
#pragma unroll 2
  for (int c = 0; c < NCORR; ++c) acc = fmaf(cp[(size_t)c * HWN], wo[c], acc);
  offs[t] = acc;  // t == ((lvl*4+b)*72+o)*4096 + hw
}

// ---------------------------------------------------------------------------
// 3) Weight transpose for coalesced A-fragment loads:
//    wT[((lvl*9 + k9)*256 + klin)*256 + cout] = w[(cout*256 + klin)*9 + k9]
// ---------------------------------------------------------------------------
__global__ void cca_wt_kernel(const float* __restrict__ w0,
                              const float* __restrict__ w1,
                              float* __restrict__ wT) {
  int t = blockIdx.x * 256 + threadIdx.x;   // 0 .. 2*9*256*256-1
  int cout = t & 255;
  int rest = t >> 8;
  int klin = rest & 255;
  rest >>= 8;
  int k9 = rest % 9;
  int lvl = rest / 9;
  const float* src = (lvl == 0) ? w0 : w1;
  wT[t] = src[((size_t)cout * NC + klin) * 9 + k9];
}

// ---------------------------------------------------------------------------
// 4) Deformable conv via fp32 WMMA implicit GEMM.
//    Block: 256 threads = 8 waves, 16-pixel tile (one row segment).
//    Loop taps(9) x groups(4): sample 64ch x 16px bilinear slab into LDS,
//    then each wave does 16 K-chunks x 2 Cout-tiles of v_wmma_f32_16x16x4_f32.
//    A (weights, from transposed wT) : lane<16 -> K rows r,r+1 in v0/v1 for
//    m=lane; lane>=16 -> rows r+2,r+3.  B (samples): v0 rows {0,2}, v1 rows
//    {1,3} across lane halves.  C/D: v_i = rows i / i+8 per lane half.
// ---------------------------------------------------------------------------
__global__ void cca_deform_kernel(const float* __restrict__ x,     // (4,256,64,64) t=1
                                  const float* __restrict__ offs,  // (4,72,64,64)
                                  const float* __restrict__ wT,    // (9,256,256)
                                  float* __restrict__ out) {       // (4,256,64,64)
  constexpr int SP = 17;  // LDS pitch (floats) to avoid bank conflicts
  __shared__ float S[64 * SP];

  const int tid = threadIdx.x;
  const int b = blockIdx.y;
  const int hw0 = blockIdx.x << 4;  // 16 consecutive pixels in one row
  const int h = hw0 >> 6;
  const int w0 = hw0 & (WW - 1);

  const int n = tid >> 4;    // pixel within tile (0..15)
  const int j = tid & 15;    // channel lane (0..15)
  const int lane = tid & 31;
  const int wave = tid >> 5;     // 0..7
  const int l = lane & 15;
  const int half = lane >> 4;    // 0 or 1
  const int cout0 = wave * 32;   // this wave's first Cout tile; second = +16

  v8f acc0 = {0.f, 0.f, 0.f, 0.f, 0.f, 0.f, 0.f, 0.f};
  v8f acc1 = {0.f, 0.f, 0.f, 0.f, 0.f, 0.f, 0.f, 0.f};

  const float* xb = x + (size_t)b * NC * HWN;
  const float* ob = offs + (size_t)b * OFFCH * HWN;
  const int hw = hw0 + n;
  const int wpix = w0 + n;

  for (int k9 = 0; k9 < 9; ++k9) {
    const int kh = k9 / 3, kw = k9 % 3;
    for (int g = 0; g < 4; ++g) {
      // ---- bilinear sampling: clamp + mask folded into weights (no EXEC
      //      branches), 4 unconditional gathers per channel ----
      float oy = ob[(size_t)(g * 18 + k9 * 2 + 0) * HWN + hw];
      float ox = ob[(size_t)(g * 18 + k9 * 2 + 1) * HWN + hw];
      float py = (float)(h + kh - 1) + oy;
      float px = (float)(wpix + kw - 1) + ox;
      float fy = floorf(py), fx = floorf(px);
      float wy = py - fy, wx = px - fx;
      int y0 = (int)fy, x0i = (int)fx;
      int y1 = y0 + 1, x1i = x0i + 1;
      float my0 = (y0 >= 0 && y0 < HH) ? 1.f : 0.f;
      float my1 = (y1 >= 0 && y1 < HH) ? 1.f : 0.f;
      float mx0 = (x0i >= 0 && x0i < WW) ? 1.f : 0.f;
      float mx1 = (x1i >= 0 && x1i < WW) ? 1.f : 0.f;
      float w00 = (1.f - wy) * (1.f - wx) * my0 * mx0;
      float w01 = (1.f - wy) * wx * my0 * mx1;
      float w10 = wy * (1.f - wx) * my1 * mx0;
      float w11 = wy * wx * my1 * mx1;
      int y0c = y0 < 0 ? 0 : (y0 > HH - 1 ? HH - 1 : y0);
      int y1c = y1 < 0 ? 0 : (y1 > HH - 1 ? HH - 1 : y1);
      int x0c = x0i < 0 ? 0 : (x0i > WW - 1 ? WW - 1 : x0i);
      int x1c = x1i < 0 ? 0 : (x1i > WW - 1 ? WW - 1 : x1i);
      int i00 = y0c * WW + x0c, i01 = y0c * WW + x1c;
      int i10 = y1c * WW + x0c, i11 = y1c * WW + x1c;
      const float* xg = xb + (size_t)g * 64 * HWN;
#pragma unroll
      for (int cj = 0; cj < 4; ++cj) {
        int c = j + cj * 16;  // channel within group, 0..63
        const float* xc = xg + (size_t)c * HWN;
        S[c * SP + n] =
            fmaf(w00, xc[i00],
                 fmaf(w01, xc[i01], fmaf(w10, xc[i10], w11 * xc[i11])));
      }
      __syncthreads();

      // ---- WMMA GEMM over this group's 64 reduction rows ----
      const int klin0 = g * 64;
#pragma unroll 4
      for (int kc = 0; kc < 16; ++kc) {
        const int r = kc * 4 + half * 2;  // lane-half selects K rows 0/1 vs 2/3
        v2f bf;
        bf.x = S[(r + 0) * SP + l];
        bf.y = S[(r + 1) * SP + l];
        // coalesced: consecutive lanes -> consecutive couts
        const float* wp = wT + (((size_t)k9 * NC + klin0 + r) << 8) + cout0 + l;
        __builtin_prefetch(wp + (4 << 8), 0, 0);
        v2f a0;
        a0.x = wp[0];
        a0.y = wp[256];
        v2f a1;
        a1.x = wp[16];
        a1.y = wp[256 + 16];
        acc0 = __builtin_amdgcn_wmma_f32_16x16x4_f32(false, a0, false, bf,
                                                     (short)0, acc0, false, false);
        acc1 = __builtin_amdgcn_wmma_f32_16x16x4_f32(false, a1, false, bf,
                                                     (short)0, acc1, false, false);
      }
      __syncthreads();
    }
  }

  // ---- write D tiles: v_i -> rows i (lanes 0-15) / i+8 (lanes 16-31) ----
#pragma unroll
  for (int i = 0; i < 8; ++i) {
    int m = i + half * 8;
    out[((size_t)(b * NC + cout0 + m)) * HWN + hw0 + l] = acc0[i];
    out[((size_t)(b * NC + cout0 + 16 + m)) * HWN + hw0 + l] = acc1[i];
  }
}

// ---------------------------------------------------------------------------
extern "C" void kernel_launch(void* const* d_in, const int* in_sizes, int n_in,
                              void* d_out, int out_size, void* d_ws,
                              size_t ws_size, hipStream_t stream) {
  const float* x0 = (const float*)d_in[0];      // (2,4,256,64,64)
  const float* x1 = (const float*)d_in[1];      // (2,4,256,64,64)
  const float* w_off0 = (const float*)d_in[2];  // (72,106)
  const float* w_off1 = (const float*)d_in[3];  // (72,106)
  const float* w_def0 = (const float*)d_in[4];  // (256,256,3,3)
  const float* w_def1 = (const float*)d_in[5];  // (256,256,3,3)
  float* out = (float*)d_out;
  float* ws = (float*)d_ws;

  const size_t BCHW = (size_t)NB * NC * HWN;   // 4,194,304 (one time slice)
  const size_t OUT_T = 2 * BCHW;               // 8,388,608 (one output tensor)

  float* corr = ws;                                        // 4*106*4096
  float* offs = corr + (size_t)NB * NCORR * HWN;           // 2*4*72*4096
  float* wT = offs + (size_t)2 * NB * OFFCH * HWN;         // 2*9*256*256

  // out[l][0:4] = inp_l[t=0] passthrough
  (void)hipMemcpyAsync(out, x0, BCHW * sizeof(float), hipMemcpyDeviceToDevice,
                       stream);
  (void)hipMemcpyAsync(out + OUT_T, x1, BCHW * sizeof(float),
                       hipMemcpyDeviceToDevice, stream);

  // weight transpose (independent of everything else)
  cca_wt_kernel<<<(2 * 9 * NC * NC) / 256, 256, 0, stream>>>(w_def0, w_def1, wT);

  // correlation: f1 = inp[t=1], f2 = inp[t=0]
  cca_corr_kernel<9, 2, 4><<<dim3(64, NB), 64, 0, stream>>>(x0 + BCHW, x0, corr, 0);
  cca_corr_kernel<5, 1, 4><<<dim3(64, NB), 64, 0, stream>>>(x1 + BCHW, x1, corr, 81);

  // offsets for both levels
  cca_offset_kernel<<<(2 * NB * OFFCH * HWN) / 256, 256, 0, stream>>>(
      corr, w_off0, w_off1, offs);

  // deformable conv (WMMA implicit GEMM), feats land after the t=0 slice
  cca_deform_kernel<<<dim3(HWN / 16, NB), 256, 0, stream>>>(
      x0 + BCHW, offs, wT, out + BCHW);
  cca_deform_kernel<<<dim3(HWN / 16, NB), 256, 0, stream>>>(
      x1 + BCHW, offs + (size_t)NB * OFFCH * HWN, wT + (size_t)9 * NC * NC,
      out + OUT_T + BCHW);
}